// VGNodeAdapter_36833639531134
// MI455X (gfx1250) — compile-verified
//
#include <hip/hip_runtime.h>
#include <hip/hip_bf16.h>

typedef __attribute__((ext_vector_type(2))) float v2f;
typedef __attribute__((ext_vector_type(8))) float v8f;

#define HB 65536
#define OUTD 128
#define BASEIN 248
#define AST 260   // padded LDS stride for 32x248 A tile
#define HST 132   // padded LDS stride for 32x128 hidden tile

__device__ __forceinline__ float gelu_f(float x) {
  return 0.5f * x * (1.0f + erff(x * 0.70710678118654752440f));
}

// A fragment for V_WMMA_F32_16X16X4_F32.
// lanes 0-15: row m = lane, {K=k, K=k+1}; lanes 16-31: row m = lane-16, {K=k+2, K=k+3}
__device__ __forceinline__ v2f a_frag(const float* base, int lane, int k, int stride) {
  int m  = lane & 15;
  int kk = k + ((lane >> 4) << 1);
  v2f a;
  a.x = base[m * stride + kk];
  a.y = base[m * stride + kk + 1];
  return a;
}

// B fragment (row-major B[k][n], leading dim ldn).
// lanes 0-15: {B[k][n0+t], B[k+1][n0+t]}; lanes 16-31: {B[k+2][n0+t], B[k+3][n0+t]}
__device__ __forceinline__ v2f b_frag(const float* B, int lane, int k, int n0, int ldn) {
  int t  = lane & 15;
  int kk = k + ((lane >> 4) << 1);
  v2f b;
  b.x = B[kk * ldn + n0 + t];
  b.y = B[(kk + 1) * ldn + n0 + t];
  return b;
}

// -------------------------------------------------------------------------
// Kernel 1: per-hash-bucket relation MLP table.
// rel_tab[h] = gelu(rel_hash_emb[h] @ rw1 + rb1) @ rw2 + rb2   (h = 0..65535)
// One wave (one block) handles 32 buckets: two 16-row tiles sharing B frags.
// -------------------------------------------------------------------------
__global__ __launch_bounds__(32) void rel_table_kernel(
    const float* __restrict__ rel_hash_emb,
    const float* __restrict__ rw1, const float* __restrict__ rb1,
    const float* __restrict__ rw2, const float* __restrict__ rb2,
    float* __restrict__ rel_tab)
{
  __shared__ float Hid[32][HST];
  int lane = threadIdx.x & 31;
  int h0   = blockIdx.x * 32;

  __builtin_prefetch(rw1, 0, 3);
  __builtin_prefetch(rw2, 0, 3);

  int half = lane >> 4;
  int tc   = lane & 15;

  v8f acc0[8] = {};
  v8f acc1[8] = {};
  // GEMM1: [32x32] @ [32x128], A direct from global
  for (int ks = 0; ks < 8; ++ks) {
    int k = ks * 4;
    v2f af0 = a_frag(rel_hash_emb + (size_t)h0 * 32, lane, k, 32);
    v2f af1 = a_frag(rel_hash_emb + (size_t)(h0 + 16) * 32, lane, k, 32);
#pragma unroll
    for (int t = 0; t < 8; ++t) {
      v2f bf = b_frag(rw1, lane, k, t * 16, OUTD);
      acc0[t] = __builtin_amdgcn_wmma_f32_16x16x4_f32(
          false, af0, false, bf, (short)0, acc0[t], false, false);
      acc1[t] = __builtin_amdgcn_wmma_f32_16x16x4_f32(
          false, af1, false, bf, (short)0, acc1[t], false, false);
    }
  }
  // bias + GELU -> hidden tile in LDS
#pragma unroll
  for (int t = 0; t < 8; ++t) {
    int col = t * 16 + tc;
    float bv = rb1[col];
#pragma unroll
    for (int r = 0; r < 8; ++r) {
      int row = r + half * 8;
      Hid[row][col]      = gelu_f(acc0[t][r] + bv);
      Hid[16 + row][col] = gelu_f(acc1[t][r] + bv);
    }
  }
  asm volatile("s_wait_dscnt 0" ::: "memory");

  // GEMM2: [32x128] @ [128x128]
#pragma unroll
  for (int t = 0; t < 8; ++t) { acc0[t] = (v8f){}; acc1[t] = (v8f){}; }
  for (int ks = 0; ks < 32; ++ks) {
    int k = ks * 4;
    v2f af0 = a_frag(&Hid[0][0], lane, k, HST);
    v2f af1 = a_frag(&Hid[16][0], lane, k, HST);
#pragma unroll
    for (int t = 0; t < 8; ++t) {
      v2f bf = b_frag(rw2, lane, k, t * 16, OUTD);
      acc0[t] = __builtin_amdgcn_wmma_f32_16x16x4_f32(
          false, af0, false, bf, (short)0, acc0[t], false, false);
      acc1[t] = __builtin_amdgcn_wmma_f32_16x16x4_f32(
          false, af1, false, bf, (short)0, acc1[t], false, false);
    }
  }
  // + rb2, store table
#pragma unroll
  for (int t = 0; t < 8; ++t) {
    int col = t * 16 + tc;
    float bv = rb2[col];
#pragma unroll
    for (int r = 0; r < 8; ++r) {
      int row = r + half * 8;
      rel_tab[(size_t)(h0 + row) * OUTD + col]      = acc0[t][r] + bv;
      rel_tab[(size_t)(h0 + 16 + row) * OUTD + col] = acc1[t][r] + bv;
    }
  }
}

// -------------------------------------------------------------------------
// Kernel 2: edge aggregation. One wave per edge: gather rel_tab row, atomic
// add into agg[dst], count degree.
// -------------------------------------------------------------------------
__global__ __launch_bounds__(256) void edge_agg_kernel(
    const int* __restrict__ edge_index,   // (2, E)
    const int* __restrict__ ehash,        // (E,)
    const float* __restrict__ rel_tab,
    float* __restrict__ agg, float* __restrict__ deg, int E)
{
  int e    = (int)((blockIdx.x * (unsigned)blockDim.x + threadIdx.x) >> 5);
  int lane = threadIdx.x & 31;
  if (e >= E) return;
  int d = edge_index[(size_t)E + e];      // dst row
  int h = ehash[e] & (HB - 1);
  const float* __restrict__ src = rel_tab + (size_t)h * OUTD;
  float* __restrict__ dp = agg + (size_t)d * OUTD;
#pragma unroll
  for (int j = 0; j < 4; ++j) {
    int c = lane * 4 + j;
    atomicAdd(&dp[c], src[c]);
  }
  if (lane == 0) atomicAdd(&deg[d], 1.0f);
}

// -------------------------------------------------------------------------
// Kernel 3: node path. One wave (one block) per 32-node tile.
// feature gathers + bbox MLP -> LDS A tile, WMMA GEMM1+GELU, WMMA GEMM2,
// add rel aggregate, LayerNorm, coalesced float4 store.
// -------------------------------------------------------------------------
__global__ __launch_bounds__(32) void node_kernel(
    const int* __restrict__ obj_id, const int* __restrict__ attr_id,
    const float* __restrict__ bbox,
    const float* __restrict__ obj_emb, const float* __restrict__ attr_emb,
    const float* __restrict__ obj_hash_emb, const float* __restrict__ attr_hash_emb,
    const float* __restrict__ bw1, const float* __restrict__ bb1,
    const float* __restrict__ bw2, const float* __restrict__ bb2,
    const float* __restrict__ pw1, const float* __restrict__ pb1,
    const float* __restrict__ pw2, const float* __restrict__ pb2,
    const float* __restrict__ agg, const float* __restrict__ deg,
    const float* __restrict__ lng, const float* __restrict__ lnb,
    float* __restrict__ out, int Nn)
{
  __shared__ float A[32][AST];
  __shared__ float Hid[32][HST];
  __shared__ float MuSig[32][2];

  int lane = threadIdx.x & 31;
  int m0   = blockIdx.x * 32;

  __builtin_prefetch(pw1, 0, 3);
  __builtin_prefetch(pw2, 0, 3);

  // ----- feature build: one lane per node -----
  int n = m0 + lane;
  bool valid = (n < Nn);
  int ni = valid ? n : m0;

  int oid = obj_id[ni];
  for (int j = 0; j < 80; ++j)
    A[lane][j] = obj_emb[oid * 80 + j];

  int   aid[8];
  float msk[8];
  float cnt = 0.0f;
#pragma unroll
  for (int k = 0; k < 8; ++k) {
    aid[k] = attr_id[(size_t)ni * 8 + k];
    msk[k] = (aid[k] != 0) ? 1.0f : 0.0f;
    cnt += msk[k];
  }
  float inv_dn = 1.0f / fmaxf(cnt, 1.0f);

  for (int j = 0; j < 56; ++j) {
    float s = 0.0f;
#pragma unroll
    for (int k = 0; k < 8; ++k) s += msk[k] * attr_emb[aid[k] * 56 + j];
    A[lane][80 + j] = s * inv_dn;
  }
  for (int j = 0; j < 32; ++j)
    A[lane][136 + j] = obj_hash_emb[(size_t)(oid & (HB - 1)) * 32 + j];
  for (int j = 0; j < 32; ++j) {
    float s = 0.0f;
#pragma unroll
    for (int k = 0; k < 8; ++k)
      s += msk[k] * attr_hash_emb[(size_t)(aid[k] & (HB - 1)) * 32 + j];
    A[lane][168 + j] = s * inv_dn;
  }

  // bbox MLP: feats(8) -> 48 -> 48, fully per-lane in registers
  {
    float bx = bbox[(size_t)ni * 4 + 0], by = bbox[(size_t)ni * 4 + 1];
    float bw = bbox[(size_t)ni * 4 + 2], bh = bbox[(size_t)ni * 4 + 3];
    float ft[8] = { bx, by, bw, bh, bx + 0.5f * bw, by + 0.5f * bh,
                    fmaxf(bw * bh, 0.0f), bw / fmaxf(bh, 1e-6f) };
    float h1[48];
    for (int c = 0; c < 48; ++c) {
      float s = bb1[c];
#pragma unroll
      for (int i = 0; i < 8; ++i) s += ft[i] * bw1[i * 48 + c];
      h1[c] = gelu_f(s);
    }
    for (int c = 0; c < 48; ++c) {
      float s = bb2[c];
      for (int i = 0; i < 48; ++i) s += h1[i] * bw2[i * 48 + c];
      A[lane][200 + c] = gelu_f(s);
    }
  }
  asm volatile("s_wait_dscnt 0" ::: "memory");

  int half = lane >> 4;
  int tc   = lane & 15;

  // ----- GEMM1: [32x248] @ pw1[248x128], two 16-row tiles share B frags -----
  v8f acc0[8] = {};
  v8f acc1[8] = {};
  for (int ks = 0; ks < 62; ++ks) {
    int k = ks * 4;
    v2f af0 = a_frag(&A[0][0], lane, k, AST);
    v2f af1 = a_frag(&A[16][0], lane, k, AST);
#pragma unroll
    for (int t = 0; t < 8; ++t) {
      v2f bf = b_frag(pw1, lane, k, t * 16, OUTD);
      acc0[t] = __builtin_amdgcn_wmma_f32_16x16x4_f32(
          false, af0, false, bf, (short)0, acc0[t], false, false);
      acc1[t] = __builtin_amdgcn_wmma_f32_16x16x4_f32(
          false, af1, false, bf, (short)0, acc1[t], false, false);
    }
  }
  // bias + GELU -> hidden tile
#pragma unroll
  for (int t = 0; t < 8; ++t) {
    int col = t * 16 + tc;
    float bv = pb1[col];
#pragma unroll
    for (int r = 0; r < 8; ++r) {
      int row = r + half * 8;
      Hid[row][col]      = gelu_f(acc0[t][r] + bv);
      Hid[16 + row][col] = gelu_f(acc1[t][r] + bv);
    }
  }
  asm volatile("s_wait_dscnt 0" ::: "memory");

  // ----- GEMM2: [32x128] @ pw2[128x128] -----
#pragma unroll
  for (int t = 0; t < 8; ++t) { acc0[t] = (v8f){}; acc1[t] = (v8f){}; }
  for (int ks = 0; ks < 32; ++ks) {
    int k = ks * 4;
    v2f af0 = a_frag(&Hid[0][0], lane, k, HST);
    v2f af1 = a_frag(&Hid[16][0], lane, k, HST);
#pragma unroll
    for (int t = 0; t < 8; ++t) {
      v2f bf = b_frag(pw2, lane, k, t * 16, OUTD);
      acc0[t] = __builtin_amdgcn_wmma_f32_16x16x4_f32(
          false, af0, false, bf, (short)0, acc0[t], false, false);
      acc1[t] = __builtin_amdgcn_wmma_f32_16x16x4_f32(
          false, af1, false, bf, (short)0, acc1[t], false, false);
    }
  }

  // ----- epilogue: + pb2 + rel aggregate, stash hsum in LDS -----
#pragma unroll
  for (int t = 0; t < 8; ++t) {
    int col = t * 16 + tc;
    float bv = pb2[col];
#pragma unroll
    for (int r = 0; r < 8; ++r) {
      int row0 = r + half * 8;
      int row1 = row0 + 16;
      int n0c  = (m0 + row0 < Nn) ? (m0 + row0) : m0;
      int n1c  = (m0 + row1 < Nn) ? (m0 + row1) : m0;
      float rv0 = agg[(size_t)n0c * OUTD + col] / fmaxf(deg[n0c], 1.0f);
      float rv1 = agg[(size_t)n1c * OUTD + col] / fmaxf(deg[n1c], 1.0f);
      Hid[row0][col] = acc0[t][r] + bv + rv0;
      Hid[row1][col] = acc1[t][r] + bv + rv1;
    }
  }
  asm volatile("s_wait_dscnt 0" ::: "memory");

  // ----- LayerNorm stats: one lane per row -----
  {
    float mu = 0.0f;
    for (int c = 0; c < OUTD; ++c) mu += Hid[lane][c];
    mu *= (1.0f / 128.0f);
    float var = 0.0f;
    for (int c = 0; c < OUTD; ++c) {
      float d = Hid[lane][c] - mu;
      var += d * d;
    }
    var *= (1.0f / 128.0f);
    MuSig[lane][0] = mu;
    MuSig[lane][1] = rsqrtf(var + 1e-5f);
  }
  asm volatile("s_wait_dscnt 0" ::: "memory");

  // ----- normalize + coalesced store: 4 cols per lane per row -----
  {
    int c0 = lane * 4;
    float lg0 = lng[c0 + 0], lg1 = lng[c0 + 1], lg2 = lng[c0 + 2], lg3 = lng[c0 + 3];
    float lb0 = lnb[c0 + 0], lb1 = lnb[c0 + 1], lb2 = lnb[c0 + 2], lb3 = lnb[c0 + 3];
    for (int r = 0; r < 32; ++r) {
      if (m0 + r >= Nn) break;
      float mu = MuSig[r][0];
      float rs = MuSig[r][1];
      float4 o;
      o.x = (Hid[r][c0 + 0] - mu) * rs * lg0 + lb0;
      o.y = (Hid[r][c0 + 1] - mu) * rs * lg1 + lb1;
      o.z = (Hid[r][c0 + 2] - mu) * rs * lg2 + lb2;
      o.w = (Hid[r][c0 + 3] - mu) * rs * lg3 + lb3;
      *(float4*)&out[(size_t)(m0 + r) * OUTD + c0] = o;
    }
  }
}

// -------------------------------------------------------------------------
extern "C" void kernel_launch(void* const* d_in, const int* in_sizes, int n_in,
                              void* d_out, int out_size, void* d_ws, size_t ws_size,
                              hipStream_t stream)
{
  const int*   obj_id        = (const int*)d_in[0];
  const int*   attr_id       = (const int*)d_in[1];
  const float* bbox          = (const float*)d_in[2];
  const int*   edge_index    = (const int*)d_in[3];
  const int*   ehash         = (const int*)d_in[4];
  const float* obj_emb       = (const float*)d_in[5];
  const float* attr_emb      = (const float*)d_in[6];
  const float* obj_hash_emb  = (const float*)d_in[7];
  const float* attr_hash_emb = (const float*)d_in[8];
  const float* rel_hash_emb  = (const float*)d_in[9];
  const float* bw1 = (const float*)d_in[10];
  const float* bb1 = (const float*)d_in[11];
  const float* bw2 = (const float*)d_in[12];
  const float* bb2 = (const float*)d_in[13];
  const float* pw1 = (const float*)d_in[14];
  const float* pb1 = (const float*)d_in[15];
  const float* pw2 = (const float*)d_in[16];
  const float* pb2 = (const float*)d_in[17];
  const float* rw1 = (const float*)d_in[18];
  const float* rb1 = (const float*)d_in[19];
  const float* rw2 = (const float*)d_in[20];
  const float* rb2 = (const float*)d_in[21];
  const float* lng = (const float*)d_in[22];
  const float* lnb = (const float*)d_in[23];

  int N = in_sizes[0];
  int E = in_sizes[4];

  float* rel_tab = (float*)d_ws;
  float* agg     = rel_tab + (size_t)HB * OUTD;
  float* deg     = agg + (size_t)N * OUTD;

  // zero agg + deg (contiguous)
  hipMemsetAsync(agg, 0, ((size_t)N * OUTD + (size_t)N) * sizeof(float), stream);

  // 1) relation table over all hash buckets: 32 buckets per single-wave block
  rel_table_kernel<<<HB / 32, 32, 0, stream>>>(rel_hash_emb, rw1, rb1, rw2, rb2, rel_tab);

  // 2) edge aggregation: one wave per edge, 8 edges per 256-thread block
  int eb = (E + 7) / 8;
  edge_agg_kernel<<<eb, 256, 0, stream>>>(edge_index, ehash, rel_tab, agg, deg, E);

  // 3) node path: 32 nodes per single-wave block
  int nb = (N + 31) / 32;
  node_kernel<<<nb, 32, 0, stream>>>(obj_id, attr_id, bbox, obj_emb, attr_emb,
                                     obj_hash_emb, attr_hash_emb,
                                     bw1, bb1, bw2, bb2, pw1, pb1, pw2, pb2,
                                     agg, deg, lng, lnb, (float*)d_out, N);
}